// MultiBoxLoss_45226005627590
// MI455X (gfx1250) — compile-verified
//
#include <hip/hip_runtime.h>
#include <hip/hip_bf16.h>

// Problem constants (from the reference)
#define B_      32
#define P_      32768
#define O_      32
#define THRESH  0.35f
#define NEGPOS  7
#define TPB     256
#define NTILE   (P_ / TPB)        // 128 prior-tiles per batch
#define NBLK    (B_ * NTILE)      // 4096 partial slots

typedef float v2f __attribute__((ext_vector_type(2)));
typedef float v8f __attribute__((ext_vector_type(8)));

// ---------------------------------------------------------------------------
// Reset integer accumulators (atomic int adds are deterministic).
// ---------------------------------------------------------------------------
__global__ void zero_npos_kernel(int* __restrict__ npos) {
    const int t = threadIdx.x;
    if (t < B_) npos[t] = 0;
}

// ---------------------------------------------------------------------------
// best_prior_idx[b,o] = argmax_p IoU(truth[b,o], point_form(prior[p]))
// One block per (b,o). Priors (512KB) stay L2-resident across all 1024 blocks.
// Tie-break: smaller p (JAX argmax = first max).
// ---------------------------------------------------------------------------
__global__ void best_prior_kernel(const float4* __restrict__ pri,
                                  const float*  __restrict__ tgt,
                                  int*          __restrict__ bpi) {
    const int o = blockIdx.x, b = blockIdx.y, t = threadIdx.x;
    const size_t tb = ((size_t)b * O_ + o) * 5;
    const float ax1 = tgt[tb + 0], ay1 = tgt[tb + 1];
    const float ax2 = tgt[tb + 2], ay2 = tgt[tb + 3];
    const float area_a = (ax2 - ax1) * (ay2 - ay1);

    float bv = -1.0f; int bi = P_;
    for (int p = t; p < P_; p += TPB) {
        __builtin_prefetch(&pri[p + 2 * TPB], 0, 0);   // global_prefetch_b8
        const float4 pr = pri[p];                      // global_load_b128
        const float bx1 = pr.x - 0.5f * pr.z, by1 = pr.y - 0.5f * pr.w;
        const float bx2 = pr.x + 0.5f * pr.z, by2 = pr.y + 0.5f * pr.w;
        const float iw = fmaxf(fminf(ax2, bx2) - fmaxf(ax1, bx1), 0.0f);
        const float ih = fmaxf(fminf(ay2, by2) - fmaxf(ay1, by1), 0.0f);
        const float inter = iw * ih;
        const float iou = inter / (area_a + pr.z * pr.w - inter);
        if (iou > bv) { bv = iou; bi = p; }            // strided => first max kept
    }

    __shared__ float sv[TPB];
    __shared__ int   si[TPB];
    sv[t] = bv; si[t] = bi; __syncthreads();
    for (int s = TPB / 2; s > 0; s >>= 1) {
        if (t < s) {
            const float v2 = sv[t + s]; const int i2 = si[t + s];
            if (v2 > sv[t] || (v2 == sv[t] && i2 < si[t])) { sv[t] = v2; si[t] = i2; }
        }
        __syncthreads();
    }
    if (t == 0) bpi[b * O_ + o] = si[0];
}

// ---------------------------------------------------------------------------
// Per-prior matching + both loss contributions. One thread per (b,p).
// Streams loc (b128) and conf (b64) exactly once; truths staged in LDS.
// Writes: mine[b,p], deterministic per-block partials, int atomic num_pos.
// ---------------------------------------------------------------------------
__global__ void match_loss_kernel(const float4* __restrict__ loc,
                                  const float2* __restrict__ conf,
                                  const float4* __restrict__ pri,
                                  const float*  __restrict__ tgt,
                                  const int*    __restrict__ bpi,
                                  float*        __restrict__ mine,
                                  int*          __restrict__ npos,
                                  float*        __restrict__ part_l,
                                  float*        __restrict__ part_c) {
    const int b = blockIdx.y, t = threadIdx.x;
    const int p = blockIdx.x * TPB + t;

    __shared__ float T[O_ * 5];
    __shared__ int   sb[O_];
    if (t < O_ * 5) T[t]  = tgt[(size_t)b * O_ * 5 + t];
    if (t < O_)     sb[t] = bpi[b * O_ + t];
    __syncthreads();

    __builtin_prefetch(&loc [(size_t)b * P_ + p], 0, 0);
    __builtin_prefetch(&conf[(size_t)b * P_ + p], 0, 0);

    const float4 pr = pri[p];
    const float bx1 = pr.x - 0.5f * pr.z, by1 = pr.y - 0.5f * pr.w;
    const float bx2 = pr.x + 0.5f * pr.z, by2 = pr.y + 0.5f * pr.w;
    const float area_b = pr.z * pr.w;

    // best truth per prior (argmax over O, first-max tie-break)
    float bv = -1.0f; int bi = 0;
#pragma unroll 4
    for (int o = 0; o < O_; ++o) {
        const float ax1 = T[o*5+0], ay1 = T[o*5+1], ax2 = T[o*5+2], ay2 = T[o*5+3];
        const float iw = fmaxf(fminf(ax2, bx2) - fmaxf(ax1, bx1), 0.0f);
        const float ih = fmaxf(fminf(ay2, by2) - fmaxf(ay1, by1), 0.0f);
        const float inter  = iw * ih;
        const float area_a = (ax2 - ax1) * (ay2 - ay1);
        const float iou = inter / (area_a + area_b - inter);
        if (iou > bv) { bv = iou; bi = o; }
    }
    // scatter "fix": forced matches from best_prior_idx (last-wins on dup)
    for (int o = 0; o < O_; ++o)
        if (sb[o] == p) { bv = 2.0f; bi = o; }

    int ct = (int)T[bi*5+4] + 1;
    if (bv < THRESH) ct = 0;
    const bool pos = ct > 0;

    // encode(matched, prior) and smooth-L1 vs loc_data
    const float mx1 = T[bi*5+0], my1 = T[bi*5+1], mx2 = T[bi*5+2], my2 = T[bi*5+3];
    const float gx = ((mx1 + mx2) * 0.5f - pr.x) / (0.1f * pr.z);
    const float gy = ((my1 + my2) * 0.5f - pr.y) / (0.1f * pr.w);
    const float gw = logf((mx2 - mx1) / pr.z) * 5.0f;   // /0.2
    const float gh = logf((my2 - my1) / pr.w) * 5.0f;

    const float4 ld = loc[(size_t)b * P_ + p];
    auto sl1 = [](float d) { const float a = fabsf(d); return a < 1.0f ? 0.5f * d * d : a - 0.5f; };
    const float l = sl1(ld.x - gx) + sl1(ld.y - gy) + sl1(ld.z - gw) + sl1(ld.w - gh);
    const float lpart = pos ? l : 0.0f;

    // 2-class cross entropy
    const float2 cf = conf[(size_t)b * P_ + p];
    const float m   = fmaxf(cf.x, cf.y);
    const float lse = m + logf(expf(cf.x - m) + expf(cf.y - m));
    const float ce  = lse - (ct == 0 ? cf.x : cf.y);
    mine[(size_t)b * P_ + p] = pos ? 0.0f : ce;
    const float cpart = pos ? ce : 0.0f;

    // deterministic block reductions -> fixed partial slots
    __shared__ float rl[TPB]; __shared__ float rc[TPB]; __shared__ int rp[TPB];
    rl[t] = lpart; rc[t] = cpart; rp[t] = pos ? 1 : 0; __syncthreads();
    for (int s = TPB / 2; s > 0; s >>= 1) {
        if (t < s) { rl[t] += rl[t+s]; rc[t] += rc[t+s]; rp[t] += rp[t+s]; }
        __syncthreads();
    }
    if (t == 0) {
        const int bid = b * gridDim.x + blockIdx.x;
        part_l[bid] = rl[0];
        part_c[bid] = rc[0];
        if (rp[0]) atomicAdd(&npos[b], rp[0]);
    }
}

// ---------------------------------------------------------------------------
// Exact per-batch top-K sum of `mine` (hard negative mining). All mine >= 0,
// so the raw float bits are a monotone uint key: 32-step binary search for the
// K-th largest, then one threshold pass with exact tie handling. All passes
// hit L2 (128KB/batch). Fixed-order reductions => deterministic.
// ---------------------------------------------------------------------------
__global__ void hardneg_kernel(const float* __restrict__ mine,
                               const int*   __restrict__ npos,
                               float*       __restrict__ lcneg) {
    const int b = blockIdx.x, t = threadIdx.x;
    const float* mb = mine + (size_t)b * P_;
    __shared__ int   si[TPB];
    __shared__ float sf[TPB];

    const int K = min(NEGPOS * npos[b], P_ - 1);
    if (K <= 0) { if (t == 0) lcneg[b] = 0.0f; return; }

    unsigned lo = 0u, hi = 0xFFFFFFFFu;   // invariant: count(bits >= lo) >= K
    while (lo < hi) {
        const unsigned mid = lo + (unsigned)((((unsigned long long)(hi - lo)) + 1ull) >> 1);
        int c = 0;
        for (int i = t; i < P_; i += TPB) {
            __builtin_prefetch(&mb[i + TPB], 0, 0);
            c += (__float_as_uint(mb[i]) >= mid) ? 1 : 0;
        }
        si[t] = c; __syncthreads();
        for (int s = TPB / 2; s > 0; s >>= 1) { if (t < s) si[t] += si[t+s]; __syncthreads(); }
        const int tot = si[0];
        __syncthreads();
        if (tot >= K) lo = mid; else hi = mid - 1;
    }
    const unsigned tbits = lo;            // bits of the K-th largest value

    int cg = 0; float sg = 0.0f;
    for (int i = t; i < P_; i += TPB) {
        const float v = mb[i];
        if (__float_as_uint(v) > tbits) { cg++; sg += v; }
    }
    si[t] = cg; sf[t] = sg; __syncthreads();
    for (int s = TPB / 2; s > 0; s >>= 1) {
        if (t < s) { si[t] += si[t+s]; sf[t] += sf[t+s]; }
        __syncthreads();
    }
    if (t == 0) lcneg[b] = sf[0] + (float)(K - si[0]) * __uint_as_float(tbits);
}

// ---------------------------------------------------------------------------
// Final deterministic reduction on one wave (EXEC all-ones, as WMMA needs).
// V_WMMA_F32_16X16X4_F32 with A = ones: D[i][j] = sum_k B[k][j] + C[i][j],
// i.e. an exact f32 64-way reduction per instruction, C-chained across the
// 4096-entry partial arrays. Rows of D are identical; VGPR0 holds rows 0 and
// 8 across the 32 lanes, so a full 32-lane fold double-counts -> * 0.5.
// ---------------------------------------------------------------------------
__global__ void finalize_kernel(const float* __restrict__ part_l,
                                const float* __restrict__ part_c,
                                const float* __restrict__ lcneg,
                                const int*   __restrict__ npos,
                                float*       __restrict__ out) {
    const int lane = threadIdx.x;        // exactly 32 threads = 1 wave
    v8f accl = {}; v8f accc = {};
    const v2f ones = {1.0f, 1.0f};
    for (int base = 0; base < NBLK; base += 64) {
        v2f bl, bc;
        bl.x = part_l[base + lane]; bl.y = part_l[base + 32 + lane];
        bc.x = part_c[base + lane]; bc.y = part_c[base + 32 + lane];
        accl = __builtin_amdgcn_wmma_f32_16x16x4_f32(false, ones, false, bl,
                                                     (short)0, accl, false, false);
        accc = __builtin_amdgcn_wmma_f32_16x16x4_f32(false, ones, false, bc,
                                                     (short)0, accc, false, false);
    }
    float rl = accl[0], rc = accc[0];
    for (int off = 16; off > 0; off >>= 1) {
        rl += __shfl_xor(rl, off, 32);
        rc += __shfl_xor(rc, off, 32);
    }
    rl *= 0.5f; rc *= 0.5f;

    if (lane == 0) {
        float cneg = 0.0f; int np = 0;
        for (int b = 0; b < B_; ++b) { cneg += lcneg[b]; np += npos[b]; }
        const float N = fmaxf((float)np, 1.0f);
        out[0] = rl / N;
        out[1] = (rc + cneg) / N;
    }
}

// ---------------------------------------------------------------------------
// Host entry. Inputs: loc_data(B,P,4) f32, conf_data(B,P,2) f32,
// priors(P,4) f32, targets(B,O,5) f32. Output: 2 f32 scalars.
// Workspace (~4.3 MB): mine | bpi | npos | part_l | part_c | lcneg.
// ---------------------------------------------------------------------------
extern "C" void kernel_launch(void* const* d_in, const int* in_sizes, int n_in,
                              void* d_out, int out_size, void* d_ws, size_t ws_size,
                              hipStream_t stream) {
    const float4* loc  = (const float4*)d_in[0];
    const float2* conf = (const float2*)d_in[1];
    const float4* pri  = (const float4*)d_in[2];
    const float*  tgt  = (const float*)d_in[3];
    float* out = (float*)d_out;

    char* ws = (char*)d_ws;
    float* mine   = (float*)ws;  ws += (size_t)B_ * P_ * sizeof(float);   // 4 MB
    int*   bpi    = (int*)ws;    ws += (size_t)B_ * O_ * sizeof(int);     // 4 KB
    int*   npos   = (int*)ws;    ws += 128;                               // 32 ints, padded
    float* part_l = (float*)ws;  ws += (size_t)NBLK * sizeof(float);      // 16 KB
    float* part_c = (float*)ws;  ws += (size_t)NBLK * sizeof(float);      // 16 KB
    float* lcneg  = (float*)ws;                                           // 128 B

    zero_npos_kernel<<<1, 32, 0, stream>>>(npos);
    best_prior_kernel<<<dim3(O_, B_), TPB, 0, stream>>>(pri, tgt, bpi);
    match_loss_kernel<<<dim3(NTILE, B_), TPB, 0, stream>>>(loc, conf, pri, tgt, bpi,
                                                           mine, npos, part_l, part_c);
    hardneg_kernel<<<B_, TPB, 0, stream>>>(mine, npos, lcneg);
    finalize_kernel<<<1, 32, 0, stream>>>(part_l, part_c, lcneg, npos, out);
}